// MedianFilter_76330158785000
// MI455X (gfx1250) — compile-verified
//
#include <hip/hip_runtime.h>

// ---------------------------------------------------------------------------
// 3x3 median filter, zero padding, f32, planes of 512x512.
// Memory-bound: >= ~100 MB HBM traffic -> ~4.3 us floor at 23.3 TB/s.
// Strategy: 4x4 output register tile per thread from a 6x6 input window,
// float4 (b128) loads/stores, Paeth 19-exchange median-of-9 network
// (compiler CSE/DCE prunes to ~17 VALU ops/pixel across the tile).
// Halo loads use clamped addresses + v_cndmask selects instead of divergent
// branches (x-guards are lane-divergent; row guards are wave-uniform and
// stay as cheap scalar branches).
// ---------------------------------------------------------------------------

#define S2(a, b)                     \
    {                                \
        float _mn = fminf(a, b);     \
        (b) = fmaxf(a, b);           \
        (a) = _mn;                   \
    }

__device__ __forceinline__ float med9(float p0, float p1, float p2,
                                      float p3, float p4, float p5,
                                      float p6, float p7, float p8) {
    // Paeth median-of-9 network (19 compare-exchanges); median ends in p4.
    S2(p1, p2); S2(p4, p5); S2(p7, p8);
    S2(p0, p1); S2(p3, p4); S2(p6, p7);
    S2(p1, p2); S2(p4, p5); S2(p7, p8);
    S2(p0, p3); S2(p5, p8); S2(p4, p7);
    S2(p3, p6); S2(p1, p4); S2(p2, p5);
    S2(p4, p7); S2(p4, p2); S2(p6, p4);
    S2(p4, p2);
    return p4;
}

__global__ __launch_bounds__(256) void median3x3_kernel(
    const float* __restrict__ in, float* __restrict__ out, int nplanes) {
    constexpr int W  = 512;
    constexpr int H  = 512;
    constexpr int GX = W / 4;  // 128 col-groups of 4
    constexpr int GY = H / 4;  // 128 row-groups of 4

    const int tid   = blockIdx.x * blockDim.x + threadIdx.x;
    const int gx    = tid & (GX - 1);
    const int gy    = (tid / GX) & (GY - 1);
    const int plane = tid / (GX * GY);
    if (plane >= nplanes) return;

    const int x4 = gx * 4;  // first output column of this tile
    const int y4 = gy * 4;  // first output row of this tile

    const float* __restrict__ p = in  + (size_t)plane * H * W;
    float* __restrict__       o = out + (size_t)plane * H * W;

    // gfx1250: kick the tile's cachelines toward the WGP early
    // (lowers to global_prefetch_b8; no LOADcnt cost).
    __builtin_prefetch(p + (size_t)y4 * W + x4, 0, 0);

    // Clamped (always in-bounds) halo addresses; zero-padding applied via
    // selects below. Lane-divergent conditions -> v_cndmask, not exec-mask.
    const bool  has_l = (x4 > 0);
    const bool  has_r = (x4 + 4 < W);
    const int   xL    = has_l ? (x4 - 1) : 0;
    const int   xR    = has_r ? (x4 + 4) : (W - 1);

    // 6x6 input window: rows y4-1 .. y4+4, cols x4-1 .. x4+4 (zero padded).
    float v[6][6];
#pragma unroll
    for (int r = 0; r < 6; ++r) {
        const int yy = y4 - 1 + r;
        if (yy < 0 || yy >= H) {  // wave-uniform branch (gy uniform per wave)
#pragma unroll
            for (int c = 0; c < 6; ++c) v[r][c] = 0.0f;
        } else {
            const float* row = p + (size_t)yy * W;
            const float4 c4  = *reinterpret_cast<const float4*>(row + x4);
            const float  lv  = row[xL];  // unconditional load, clamped addr
            const float  rv  = row[xR];  // unconditional load, clamped addr
            v[r][1] = c4.x;
            v[r][2] = c4.y;
            v[r][3] = c4.z;
            v[r][4] = c4.w;
            v[r][0] = has_l ? lv : 0.0f;  // v_cndmask
            v[r][5] = has_r ? rv : 0.0f;  // v_cndmask
        }
    }

    // 4x4 outputs; output row y4+j uses window rows j, j+1, j+2.
#pragma unroll
    for (int j = 0; j < 4; ++j) {
        float4 res;
        res.x = med9(v[j + 0][0], v[j + 0][1], v[j + 0][2],
                     v[j + 1][0], v[j + 1][1], v[j + 1][2],
                     v[j + 2][0], v[j + 2][1], v[j + 2][2]);
        res.y = med9(v[j + 0][1], v[j + 0][2], v[j + 0][3],
                     v[j + 1][1], v[j + 1][2], v[j + 1][3],
                     v[j + 2][1], v[j + 2][2], v[j + 2][3]);
        res.z = med9(v[j + 0][2], v[j + 0][3], v[j + 0][4],
                     v[j + 1][2], v[j + 1][3], v[j + 1][4],
                     v[j + 2][2], v[j + 2][3], v[j + 2][4]);
        res.w = med9(v[j + 0][3], v[j + 0][4], v[j + 0][5],
                     v[j + 1][3], v[j + 1][4], v[j + 1][5],
                     v[j + 2][3], v[j + 2][4], v[j + 2][5]);
        *reinterpret_cast<float4*>(o + (size_t)(y4 + j) * W + x4) = res;
    }
}

extern "C" void kernel_launch(void* const* d_in, const int* in_sizes, int n_in,
                              void* d_out, int out_size, void* d_ws, size_t ws_size,
                              hipStream_t stream) {
    (void)n_in; (void)out_size; (void)d_ws; (void)ws_size;

    const float* in = (const float*)d_in[0];
    float* out      = (float*)d_out;

    // Planes of 512x512; reference is fixed at kernel_size == 3.
    const int nplanes  = in_sizes[0] / (512 * 512);           // 16*3 = 48
    const int nthreads = nplanes * (512 / 4) * (512 / 4);     // one thread per 4x4 tile
    const int block    = 256;                                 // 8 waves (wave32)
    const int grid     = (nthreads + block - 1) / block;

    median3x3_kernel<<<grid, block, 0, stream>>>(in, out, nplanes);
}